// MTFNet_55542517072419
// MI455X (gfx1250) — compile-verified
//
#include <hip/hip_runtime.h>
#include <hip/hip_bf16.h>
#include <math.h>

#define B_   4
#define C_   8
#define HID  128
#define H_   1024
#define W_   1024
#define OH_  256
#define OW_  256
#define NK   41      // taps
#define KPAD 104     // K padded to 26 chunks of 4
#define NCHUNK 26

typedef __attribute__((ext_vector_type(2))) float v2f;
typedef __attribute__((ext_vector_type(8))) float v8f;

// -------- Stage 0: MLP -> gains -> separable normalized taps + 2D kernel out ---
__global__ __launch_bounds__(128) void mtf_mlp_kernel(
    const float* __restrict__ G,  const float* __restrict__ W1,
    const float* __restrict__ b1, const float* __restrict__ W2,
    const float* __restrict__ b2, float* __restrict__ gn_out,
    float* __restrict__ k2_out)
{
    __shared__ float sh[HID];
    __shared__ float sgn[C_][NK];
    int tid = threadIdx.x;

    float acc = b1[tid];
    #pragma unroll
    for (int c = 0; c < C_; ++c) acc += W1[tid * C_ + c] * G[c];
    sh[tid] = (acc > 0.f) ? acc : 0.01f * acc;   // leaky_relu
    __syncthreads();

    if (tid < C_) {
        float a = b2[tid];
        for (int j = 0; j < HID; ++j) a += W2[tid * HID + j] * sh[j];
        float gain = 1.f / (1.f + expf(-a));     // sigmoid
        const float pi = 3.14159265358979323846f;
        const float fN = 0.125f;                 // 1/(2*RATIO)
        float sigma2 = -logf(gain) / (2.f * pi * pi * fN * fN);
        float tmp[NK];
        float s = 0.f;
        for (int i = 0; i < NK; ++i) {
            float t = (float)i - 20.f;
            float v = expf(-(t * t) / (2.f * sigma2));
            tmp[i] = v; s += v;
        }
        float inv = 1.f / s;                     // k2 norm = (sum g)^2, separable
        for (int i = 0; i < NK; ++i) {
            float g = tmp[i] * inv;
            sgn[tid][i] = g;
            gn_out[tid * NK + i] = g;
        }
    }
    __syncthreads();

    for (int idx = tid; idx < C_ * NK * NK; idx += HID) {
        int c = idx / (NK * NK);
        int r = idx % (NK * NK);
        k2_out[idx] = sgn[c][r / NK] * sgn[c][r % NK];
    }
}

// banded-tap fetch: gn[i] if 0<=i<41 else 0 (LDS rows padded to 44)
__device__ __forceinline__ float gtap(const float (*s_gn)[44], int ch, int i)
{
    int ic = i < 0 ? 0 : (i > 43 ? 43 : i);
    float v = s_gn[ch][ic];
    return ((unsigned)i < 41u) ? v : 0.f;
}

// -------- Stage 1: horizontal stride-4 conv as WMMA: T = X (16xK) @ Gh (Kx16) --
__global__ __launch_bounds__(256) void mtf_hconv_wmma(
    const float* __restrict__ x, const float* __restrict__ gn,
    float* __restrict__ t)
{
    __shared__ float s_x[8][16][105];   // pad 104->105: odd stride, no bank conflicts
    __shared__ float s_gn[C_][44];
    int tid  = threadIdx.x;
    int wave = tid >> 5;
    int lane = tid & 31;

    for (int idx = tid; idx < C_ * 44; idx += 256) {
        int c = idx / 44, i = idx % 44;
        s_gn[c][i] = (i < NK) ? gn[c * NK + i] : 0.f;
    }

    // 32768 tiles: bc[5] | ty[6] | tx[4] ; one wave per 16(y) x 16(ox) tile of t
    int gw  = blockIdx.x * 8 + wave;
    int bc  = gw >> 10;
    int ty  = (gw >> 4) & 63;
    int tx  = gw & 15;
    int y0  = ty * 16;
    int ox0 = tx * 16;
    int cbase = 4 * ox0 - 18;           // input col = 4*ox + 2 - 20 + i
    const float* xp = x + (size_t)bc * ((size_t)H_ * W_);

    // stage X tile [16 rows x 104 cols] into LDS, replicate-pad via clamp
    for (int idx = lane; idx < 16 * KPAD; idx += 32) {
        int r = idx / KPAD, cc = idx % KPAD;
        int col = cbase + cc;
        col = col < 0 ? 0 : (col > W_ - 1 ? W_ - 1 : col);
        s_x[wave][r][cc] = xp[(size_t)(y0 + r) * W_ + col];
    }
    __syncthreads();

    int ch    = bc & 7;
    int m     = lane & 15;              // A: row M / B: col N
    int khalf = (lane >> 4) << 1;       // k = {0,1} lanes 0-15, {2,3} lanes 16-31
    v8f acc = {};
    #pragma unroll
    for (int q = 0; q < NCHUNK; ++q) {
        int kk = 4 * q + khalf;
        v2f a, b;
        a.x = s_x[wave][m][kk];
        a.y = s_x[wave][m][kk + 1];
        int i0 = kk - 4 * m;            // Gh[k][n] = gn[k - 4n]
        b.x = gtap(s_gn, ch, i0);
        b.y = gtap(s_gn, ch, i0 + 1);
        acc = __builtin_amdgcn_wmma_f32_16x16x4_f32(
            false, a, false, b, (short)0, acc, false, false);
    }

    // D layout: VGPR r -> rows r (lanes 0-15) / r+8 (lanes 16-31), col = lane&15
    float* tp = t + (size_t)bc * ((size_t)H_ * OW_);
    int col  = ox0 + m;
    int rofs = y0 + ((lane >> 4) << 3);
    #pragma unroll
    for (int r = 0; r < 8; ++r)
        tp[(size_t)(rofs + r) * OW_ + col] = acc[r];
}

// -------- Stage 2: vertical stride-4 conv as WMMA: OUT = Gv (16xK) @ T (Kx16) --
__global__ __launch_bounds__(256) void mtf_vconv_wmma(
    const float* __restrict__ t, const float* __restrict__ gn,
    float* __restrict__ out)
{
    __shared__ float s_t[8][KPAD][16];  // half-waves hit disjoint bank groups
    __shared__ float s_gn[C_][44];
    int tid  = threadIdx.x;
    int wave = tid >> 5;
    int lane = tid & 31;

    for (int idx = tid; idx < C_ * 44; idx += 256) {
        int c = idx / 44, i = idx % 44;
        s_gn[c][i] = (i < NK) ? gn[c * NK + i] : 0.f;
    }

    // 8192 tiles: bc[5] | ty[4] | tx[4] ; one wave per 16x16 output tile
    int gw  = blockIdx.x * 8 + wave;
    int bc  = gw >> 8;
    int ty  = (gw >> 4) & 15;
    int tx  = gw & 15;
    int oy0 = ty * 16;
    int ox0 = tx * 16;
    int rbase = 4 * oy0 - 18;           // t row = 4*oy + 2 - 20 + j
    const float* tp = t + (size_t)bc * ((size_t)H_ * OW_);

    // stage T tile [104 rows x 16 cols] into LDS, replicate rows via clamp
    for (int idx = lane; idx < KPAD * 16; idx += 32) {
        int r = idx / 16, cc = idx % 16;
        int row = rbase + r;
        row = row < 0 ? 0 : (row > H_ - 1 ? H_ - 1 : row);
        s_t[wave][r][cc] = tp[(size_t)row * OW_ + (ox0 + cc)];
    }
    __syncthreads();

    int ch    = bc & 7;
    int m     = lane & 15;
    int khalf = (lane >> 4) << 1;
    v8f acc = {};
    #pragma unroll
    for (int q = 0; q < NCHUNK; ++q) {
        int kk = 4 * q + khalf;
        v2f a, b;
        int i0 = kk - 4 * m;            // Gv[m][k] = gn[k - 4m]
        a.x = gtap(s_gn, ch, i0);
        a.y = gtap(s_gn, ch, i0 + 1);
        b.x = s_t[wave][kk][m];
        b.y = s_t[wave][kk + 1][m];
        acc = __builtin_amdgcn_wmma_f32_16x16x4_f32(
            false, a, false, b, (short)0, acc, false, false);
    }

    float* op = out + (size_t)bc * ((size_t)OH_ * OW_);
    int col  = ox0 + m;
    int rofs = oy0 + ((lane >> 4) << 3);
    #pragma unroll
    for (int r = 0; r < 8; ++r)
        op[(size_t)(rofs + r) * OW_ + col] = acc[r];
}

extern "C" void kernel_launch(void* const* d_in, const int* in_sizes, int n_in,
                              void* d_out, int out_size, void* d_ws, size_t ws_size,
                              hipStream_t stream)
{
    (void)in_sizes; (void)n_in; (void)out_size; (void)ws_size;
    const float* x  = (const float*)d_in[0];   // [4,8,1024,1024]
    const float* G  = (const float*)d_in[1];   // [8]
    const float* W1 = (const float*)d_in[2];   // [128,8]
    const float* b1 = (const float*)d_in[3];   // [128]
    const float* W2 = (const float*)d_in[4];   // [8,128]
    const float* b2 = (const float*)d_in[5];   // [8]

    float* out   = (float*)d_out;                          // [4,8,256,256]
    float* k2out = out + (size_t)B_ * C_ * OH_ * OW_;      // [8,1,41,41]
    float* gnbuf = (float*)d_ws;                           // 8*41 taps
    float* tbuf  = gnbuf + 512;                            // [32,1024,256] f32

    mtf_mlp_kernel<<<1, 128, 0, stream>>>(G, W1, b1, W2, b2, gnbuf, k2out);
    mtf_hconv_wmma<<<4096, 256, 0, stream>>>(x, gnbuf, tbuf);
    mtf_vconv_wmma<<<1024, 256, 0, stream>>>(tbuf, gnbuf, out);
}